// SE3RefinementBlock_62680752718032
// MI455X (gfx1250) — compile-verified
//
#include <hip/hip_runtime.h>
#include <hip/hip_bf16.h>
#include <stdint.h>

// ---------------- constants (match reference) ----------------
constexpr int NB  = 2;     // batch
constexpr int NN  = 512;   // sequence
constexpr int ND  = 256;   // model dim
constexpr int NH  = 8;     // heads
constexpr int NHD = 32;    // head dim
constexpr int NFF = 1024;  // ffn dim
constexpr float STEP = 0.25f;

typedef __attribute__((ext_vector_type(2))) float v2f;
typedef __attribute__((ext_vector_type(8))) float v8f;

__device__ __forceinline__ float silu_f(float x) { return x / (1.f + __expf(-x)); }
__device__ __forceinline__ float sigmoid_f(float x) { return 1.f / (1.f + __expf(-x)); }

__device__ __forceinline__ v8f wmma4(v2f a, v2f b, v8f c) {
    return __builtin_amdgcn_wmma_f32_16x16x4_f32(false, a, false, b, (short)0, c, false, false);
}

// =============================================================
// LayerNorm: one block per row (B*N rows), 256 threads (== D)
// =============================================================
__global__ __launch_bounds__(256)
void ln_kernel(const float* __restrict__ x, const float* __restrict__ w,
               const float* __restrict__ bia, float* __restrict__ out)
{
    __shared__ float red[256];
    const int row = blockIdx.x, t = threadIdx.x;
    const float v = x[(size_t)row * ND + t];
    red[t] = v; __syncthreads();
    for (int s = 128; s > 0; s >>= 1) { if (t < s) red[t] += red[t + s]; __syncthreads(); }
    const float mean = red[0] * (1.0f / ND);
    __syncthreads();
    const float c = v - mean;
    red[t] = c * c; __syncthreads();
    for (int s = 128; s > 0; s >>= 1) { if (t < s) red[t] += red[t + s]; __syncthreads(); }
    const float inv = rsqrtf(red[0] * (1.0f / ND) + 1e-5f);
    out[(size_t)row * ND + t] = c * inv * w[t] + bia[t];
}

// =============================================================
// Epilogue store for one 16x16 tile of a WMMA f32 accumulator.
// OUT_MODE 0: row-major [M x NC];  OUT_MODE 1: scatter to [B,H,N,HD]
// =============================================================
template<int NC, int ACT, int OUT_MODE>
__device__ __forceinline__
void store_tile(v8f acc, int i0, int colbase, int l16, int half,
                const float* __restrict__ bias, const float* __restrict__ resid,
                float* __restrict__ out)
{
    const int col = colbase + l16;
    const float bb = bias ? bias[col] : 0.f;
#pragma unroll
    for (int r = 0; r < 8; ++r) {
        const int row = i0 + r + 8 * half;
        float v = acc[r] + bb;
        if (ACT) v = silu_f(v);
        if (resid) v += resid[(size_t)row * NC + col];
        if (OUT_MODE == 0) {
            out[(size_t)row * NC + col] = v;
        } else {
            const int b_ = row >> 9, n = row & 511;   // row -> (b, n)
            const int h_ = col >> 5, hd = col & 31;   // col -> (h, hd)
            out[(((size_t)(b_ * NH + h_) * NN + n) * NHD) + hd] = v;
        }
    }
}

// =============================================================
// fp32 WMMA GEMM, compile-time shapes. One wave computes a 16x64
// output strip (4 accumulators, 4x A-fragment reuse).
// out = act(A[Mx K] @ W[K x NC] + bias) (+ resid)
// =============================================================
template<int K, int NC, int ACT, int OUT_MODE>
__global__ __launch_bounds__(32)
void gemm_wmma(const float* __restrict__ A, const float* __restrict__ W,
               const float* __restrict__ bias, const float* __restrict__ resid,
               float* __restrict__ out)
{
    const int lane = threadIdx.x & 31;
    const int l16  = lane & 15;
    const int half = lane >> 4;
    const int j0   = blockIdx.x * 64;   // output column strip
    const int i0   = blockIdx.y * 16;   // output row tile

    v8f acc0 = {}, acc1 = {}, acc2 = {}, acc3 = {};
    const float* __restrict__ Arow = A + (size_t)(i0 + l16) * K + 2 * half;
    const float* __restrict__ Wp   = W + (size_t)(2 * half) * NC + j0 + l16;

#pragma unroll 4
    for (int k0 = 0; k0 < K; k0 += 4) {
        v2f a;
        a.x = Arow[k0];
        a.y = Arow[k0 + 1];
        const float* __restrict__ wr0 = Wp + (size_t)k0 * NC;
        const float* __restrict__ wr1 = wr0 + NC;
        v2f b0, b1, b2, b3;
        b0.x = wr0[0];  b0.y = wr1[0];
        b1.x = wr0[16]; b1.y = wr1[16];
        b2.x = wr0[32]; b2.y = wr1[32];
        b3.x = wr0[48]; b3.y = wr1[48];
        acc0 = wmma4(a, b0, acc0);
        acc1 = wmma4(a, b1, acc1);
        acc2 = wmma4(a, b2, acc2);
        acc3 = wmma4(a, b3, acc3);
    }

    store_tile<NC, ACT, OUT_MODE>(acc0, i0, j0 +  0, l16, half, bias, resid, out);
    store_tile<NC, ACT, OUT_MODE>(acc1, i0, j0 + 16, l16, half, bias, resid, out);
    store_tile<NC, ACT, OUT_MODE>(acc2, i0, j0 + 32, l16, half, bias, resid, out);
    store_tile<NC, ACT, OUT_MODE>(acc3, i0, j0 + 48, l16, half, bias, resid, out);
}

// =============================================================
// Distance-bias MLP: pb[b,h,i,j] = silu(dist*w1+b1) @ w2 + b2
// One lane per (b,i,j) pair; writes initial values into logits buffer.
// =============================================================
__global__ __launch_bounds__(256)
void dist_bias_kernel(const float* __restrict__ coords,
                      const float* __restrict__ w1, const float* __restrict__ b1,
                      const float* __restrict__ w2, const float* __restrict__ b2,
                      float* __restrict__ logits)
{
    __shared__ float sw1[ND], sb1[ND], sw2[ND * NH];
    const int t = threadIdx.x;
    sw1[t] = w1[t];
    sb1[t] = b1[t];
#pragma unroll
    for (int r = 0; r < NH; ++r) sw2[r * 256 + t] = w2[r * 256 + t];
    __syncthreads();

    const size_t gid = (size_t)blockIdx.x * 256 + t;   // over B*N*N
    const int j = (int)(gid % NN);
    const int i = (int)((gid / NN) % NN);
    const int b = (int)(gid / ((size_t)NN * NN));

    const float xi0 = coords[(size_t)(b * NN + i) * 3 + 0];
    const float xi1 = coords[(size_t)(b * NN + i) * 3 + 1];
    const float xi2 = coords[(size_t)(b * NN + i) * 3 + 2];
    const float dx = coords[(size_t)(b * NN + j) * 3 + 0] - xi0;
    const float dy = coords[(size_t)(b * NN + j) * 3 + 1] - xi1;
    const float dz = coords[(size_t)(b * NN + j) * 3 + 2] - xi2;
    const float dist = fmaxf(sqrtf(dx * dx + dy * dy + dz * dz), 1e-6f);

    float acc[NH];
#pragma unroll
    for (int h = 0; h < NH; ++h) acc[h] = b2[h];

    for (int d = 0; d < ND; ++d) {
        const float s  = dist * sw1[d] + sb1[d];
        const float tv = silu_f(s);
#pragma unroll
        for (int h = 0; h < NH; ++h) acc[h] += tv * sw2[d * NH + h];
    }
#pragma unroll
    for (int h = 0; h < NH; ++h)
        logits[((size_t)(b * NH + h) * NN + i) * NN + j] = acc[h];
}

// =============================================================
// Attention logits: logits += (q @ k^T) / sqrt(HD); apply key mask.
// One wave per 16x64 (i,j) strip, K = HD = 32 (A reused x4).
// =============================================================
__global__ __launch_bounds__(32)
void qk_logits_kernel(const float* __restrict__ q, const float* __restrict__ kbuf,
                      const unsigned char* __restrict__ mask, float* __restrict__ logits)
{
    const int lane = threadIdx.x & 31;
    const int l16 = lane & 15, half = lane >> 4;
    const int j0 = blockIdx.x * 64, i0 = blockIdx.y * 16, bh = blockIdx.z;
    const int b = bh >> 3;
    const float* __restrict__ qrow = q + ((size_t)bh * NN + (i0 + l16)) * NHD + 2 * half;
    const float* __restrict__ kb   = kbuf + (size_t)bh * NN * NHD + 2 * half;

    v8f acc0 = {}, acc1 = {}, acc2 = {}, acc3 = {};
#pragma unroll
    for (int k0 = 0; k0 < NHD; k0 += 4) {
        v2f a;
        a.x = qrow[k0];
        a.y = qrow[k0 + 1];
        v2f b0, b1, b2, b3;
        b0.x = kb[(size_t)(j0 +  0 + l16) * NHD + k0];
        b0.y = kb[(size_t)(j0 +  0 + l16) * NHD + k0 + 1];
        b1.x = kb[(size_t)(j0 + 16 + l16) * NHD + k0];
        b1.y = kb[(size_t)(j0 + 16 + l16) * NHD + k0 + 1];
        b2.x = kb[(size_t)(j0 + 32 + l16) * NHD + k0];
        b2.y = kb[(size_t)(j0 + 32 + l16) * NHD + k0 + 1];
        b3.x = kb[(size_t)(j0 + 48 + l16) * NHD + k0];
        b3.y = kb[(size_t)(j0 + 48 + l16) * NHD + k0 + 1];
        acc0 = wmma4(a, b0, acc0);
        acc1 = wmma4(a, b1, acc1);
        acc2 = wmma4(a, b2, acc2);
        acc3 = wmma4(a, b3, acc3);
    }

    float* __restrict__ Lp = logits + (size_t)bh * NN * NN;
    const float scale = 0.17677669529663687f; // 1/sqrt(32)
    v8f accs[4] = {acc0, acc1, acc2, acc3};
#pragma unroll
    for (int c = 0; c < 4; ++c) {
        const int col = j0 + c * 16 + l16;
        const bool mj = mask[b * NN + col] != 0;
#pragma unroll
        for (int r = 0; r < 8; ++r) {
            const int row = i0 + r + 8 * half;
            const float val = accs[c][r] * scale + Lp[(size_t)row * NN + col];
            Lp[(size_t)row * NN + col] = mj ? val : -10000.0f;
        }
    }
}

// =============================================================
// Row softmax + mask renorm (in place). One block per (b,h,i) row.
// =============================================================
__global__ __launch_bounds__(256)
void softmax_kernel(float* __restrict__ logits, const unsigned char* __restrict__ mask)
{
    __shared__ float red[256];
    const int row = blockIdx.x;          // bh*N + i
    const int bh = row / NN;
    const int b = bh >> 3;
    float* __restrict__ Lp = logits + (size_t)row * NN;
    const int t = threadIdx.x;

    const float x0 = Lp[t], x1 = Lp[t + 256];
    red[t] = fmaxf(x0, x1); __syncthreads();
    for (int s = 128; s > 0; s >>= 1) { if (t < s) red[t] = fmaxf(red[t], red[t + s]); __syncthreads(); }
    const float m = red[0]; __syncthreads();

    const float e0 = __expf(x0 - m), e1 = __expf(x1 - m);
    red[t] = e0 + e1; __syncthreads();
    for (int s = 128; s > 0; s >>= 1) { if (t < s) red[t] += red[t + s]; __syncthreads(); }
    const float s1 = red[0]; __syncthreads();

    float p0 = e0 / s1, p1 = e1 / s1;
    p0 *= (mask[b * NN + t]       != 0) ? 1.f : 0.f;
    p1 *= (mask[b * NN + t + 256] != 0) ? 1.f : 0.f;
    red[t] = p0 + p1; __syncthreads();
    for (int s = 128; s > 0; s >>= 1) { if (t < s) red[t] += red[t + s]; __syncthreads(); }
    const float s2 = fmaxf(red[0], 1e-6f);
    Lp[t] = p0 / s2;
    Lp[t + 256] = p1 / s2;
}

// =============================================================
// msg = attn @ v  -> scatter to [B,N,D] (heads interleaved back)
// One wave per 16x32 strip (full head dim); K = N = 512.
// =============================================================
__global__ __launch_bounds__(32)
void attn_v_kernel(const float* __restrict__ attn, const float* __restrict__ v,
                   float* __restrict__ msg)
{
    const int lane = threadIdx.x & 31;
    const int l16 = lane & 15, half = lane >> 4;
    const int i0 = blockIdx.y * 16;  // n tile
    const int bh = blockIdx.z;
    const float* __restrict__ Arow = attn + ((size_t)bh * NN + (i0 + l16)) * NN + 2 * half;
    const float* __restrict__ Vp   = v + (size_t)bh * NN * NHD + (size_t)(2 * half) * NHD + l16;

    v8f acc0 = {}, acc1 = {};
#pragma unroll 4
    for (int k0 = 0; k0 < NN; k0 += 4) {
        v2f a;
        a.x = Arow[k0];
        a.y = Arow[k0 + 1];
        const float* __restrict__ vr0 = Vp + (size_t)k0 * NHD;
        const float* __restrict__ vr1 = vr0 + NHD;
        v2f b0, b1;
        b0.x = vr0[0];  b0.y = vr1[0];
        b1.x = vr0[16]; b1.y = vr1[16];
        acc0 = wmma4(a, b0, acc0);
        acc1 = wmma4(a, b1, acc1);
    }

    const int b_ = bh >> 3, h_ = bh & 7;
    v8f accs[2] = {acc0, acc1};
#pragma unroll
    for (int c = 0; c < 2; ++c) {
#pragma unroll
        for (int r = 0; r < 8; ++r) {
            const int row = i0 + r + 8 * half;
            const int hd  = c * 16 + l16;
            msg[(size_t)(b_ * NN + row) * ND + h_ * NHD + hd] = accs[c][r];
        }
    }
}

// =============================================================
// Finalize: gate = sigmoid(g1 . cg2_w + cg2_b);
// coord_delta = sum_j mean_h(attn) * (x_j - x_i);
// coords_out = coords + STEP * gate * coord_delta
// One block per (b,i).
// =============================================================
__global__ __launch_bounds__(256)
void finalize_kernel(const float* __restrict__ g1, const float* __restrict__ cg2w,
                     const float* __restrict__ cg2b, const float* __restrict__ attn,
                     const float* __restrict__ coords, float* __restrict__ coords_out)
{
    __shared__ float red[256];
    const int row = blockIdx.x;          // b*N + i
    const int b = row >> 9, i = row & 511;
    const int t = threadIdx.x;

    // gate dot product
    red[t] = g1[(size_t)row * ND + t] * cg2w[t];
    __syncthreads();
    for (int s = 128; s > 0; s >>= 1) { if (t < s) red[t] += red[t + s]; __syncthreads(); }
    const float gate = sigmoid_f(red[0] + cg2b[0]);
    __syncthreads();

    const float xi0 = coords[(size_t)row * 3 + 0];
    const float xi1 = coords[(size_t)row * 3 + 1];
    const float xi2 = coords[(size_t)row * 3 + 2];

    float d0 = 0.f, d1 = 0.f, d2 = 0.f;
    for (int jj = t; jj < NN; jj += 256) {
        float ma = 0.f;
#pragma unroll
        for (int h = 0; h < NH; ++h)
            ma += attn[((size_t)(b * NH + h) * NN + i) * NN + jj];
        ma *= 0.125f;
        d0 += ma * (coords[(size_t)(b * NN + jj) * 3 + 0] - xi0);
        d1 += ma * (coords[(size_t)(b * NN + jj) * 3 + 1] - xi1);
        d2 += ma * (coords[(size_t)(b * NN + jj) * 3 + 2] - xi2);
    }
    red[t] = d0; __syncthreads();
    for (int s = 128; s > 0; s >>= 1) { if (t < s) red[t] += red[t + s]; __syncthreads(); }
    const float D0 = red[0]; __syncthreads();
    red[t] = d1; __syncthreads();
    for (int s = 128; s > 0; s >>= 1) { if (t < s) red[t] += red[t + s]; __syncthreads(); }
    const float D1 = red[0]; __syncthreads();
    red[t] = d2; __syncthreads();
    for (int s = 128; s > 0; s >>= 1) { if (t < s) red[t] += red[t + s]; __syncthreads(); }
    const float D2 = red[0];

    if (t == 0) {
        coords_out[(size_t)row * 3 + 0] = xi0 + STEP * gate * D0;
        coords_out[(size_t)row * 3 + 1] = xi1 + STEP * gate * D1;
        coords_out[(size_t)row * 3 + 2] = xi2 + STEP * gate * D2;
    }
}

// =============================================================
// Launch
// =============================================================
extern "C" void kernel_launch(void* const* d_in, const int* in_sizes, int n_in,
                              void* d_out, int out_size, void* d_ws, size_t ws_size,
                              hipStream_t stream)
{
    (void)in_sizes; (void)n_in; (void)out_size; (void)ws_size;

    const float* h      = (const float*)d_in[0];
    const float* coords = (const float*)d_in[1];
    const unsigned char* mask = (const unsigned char*)d_in[2];
    const float* ln1_w = (const float*)d_in[3];
    const float* ln1_b = (const float*)d_in[4];
    const float* ln2_w = (const float*)d_in[5];
    const float* ln2_b = (const float*)d_in[6];
    const float* Wq = (const float*)d_in[7];   const float* bq = (const float*)d_in[8];
    const float* Wk = (const float*)d_in[9];   const float* bk = (const float*)d_in[10];
    const float* Wv = (const float*)d_in[11];  const float* bv = (const float*)d_in[12];
    const float* Wo = (const float*)d_in[13];  const float* bo = (const float*)d_in[14];
    const float* db1_w = (const float*)d_in[15]; const float* db1_b = (const float*)d_in[16];
    const float* db2_w = (const float*)d_in[17]; const float* db2_b = (const float*)d_in[18];
    const float* cg1_w = (const float*)d_in[19]; const float* cg1_b = (const float*)d_in[20];
    const float* cg2_w = (const float*)d_in[21]; const float* cg2_b = (const float*)d_in[22];
    const float* ff1_w = (const float*)d_in[23]; const float* ff1_b = (const float*)d_in[24];
    const float* ff2_w = (const float*)d_in[25]; const float* ff2_b = (const float*)d_in[26];

    float* out        = (float*)d_out;                 // h3 [B,N,D]
    float* coords_out = out + (size_t)NB * NN * ND;    // coords [B,N,3]

    // workspace layout (floats); total 7,340,032 floats = 28 MiB
    float* wsf    = (float*)d_ws;
    float* hn     = wsf;                       // 262144
    float* qb     = wsf + 262144;              // 262144 [B,H,N,HD]
    float* kb     = wsf + 524288;              // 262144
    float* vb     = wsf + 786432;              // 262144
    float* msg    = wsf + 1048576;             // 262144 [B,N,D]
    float* h2     = wsf + 1310720;             // 262144
    float* g1     = wsf + 1572864;             // 262144
    float* hn2    = wsf + 1835008;             // 262144
    float* ffmid  = wsf + 2097152;             // 1048576 [B*N, 4D]
    float* logits = wsf + 3145728;             // 4194304 [B,H,N,N]

    const int M = NB * NN;   // 1024 rows

    // 1) LN1
    ln_kernel<<<M, 256, 0, stream>>>(h, ln1_w, ln1_b, hn);

    // 2) Q, K, V projections (WMMA), scatter to [B,H,N,HD]
    gemm_wmma<ND, ND, 0, 1><<<dim3(ND / 64, M / 16), 32, 0, stream>>>(hn, Wq, bq, nullptr, qb);
    gemm_wmma<ND, ND, 0, 1><<<dim3(ND / 64, M / 16), 32, 0, stream>>>(hn, Wk, bk, nullptr, kb);
    gemm_wmma<ND, ND, 0, 1><<<dim3(ND / 64, M / 16), 32, 0, stream>>>(hn, Wv, bv, nullptr, vb);

    // 3) distance-bias MLP -> logits buffer (initial value)
    dist_bias_kernel<<<(NB * NN * NN) / 256, 256, 0, stream>>>(coords, db1_w, db1_b,
                                                               db2_w, db2_b, logits);

    // 4) logits += qk^T / sqrt(HD), apply key mask
    qk_logits_kernel<<<dim3(NN / 64, NN / 16, NB * NH), 32, 0, stream>>>(qb, kb, mask, logits);

    // 5) softmax + mask renorm (in place -> attn)
    softmax_kernel<<<NB * NH * NN, 256, 0, stream>>>(logits, mask);

    // 6) msg = attn @ v
    attn_v_kernel<<<dim3(1, NN / 16, NB * NH), 32, 0, stream>>>(logits, vb, msg);

    // 7) h2 = h + msg @ Wo + bo
    gemm_wmma<ND, ND, 0, 0><<<dim3(ND / 64, M / 16), 32, 0, stream>>>(msg, Wo, bo, h, h2);

    // 8) g1 = silu(msg @ cg1_w + cg1_b)
    gemm_wmma<ND, ND, 1, 0><<<dim3(ND / 64, M / 16), 32, 0, stream>>>(msg, cg1_w, cg1_b, nullptr, g1);

    // 9) LN2(h2)
    ln_kernel<<<M, 256, 0, stream>>>(h2, ln2_w, ln2_b, hn2);

    // 10) ffmid = silu(hn2 @ ff1_w + ff1_b)
    gemm_wmma<ND, NFF, 1, 0><<<dim3(NFF / 64, M / 16), 32, 0, stream>>>(hn2, ff1_w, ff1_b, nullptr, ffmid);

    // 11) h3 = h2 + ffmid @ ff2_w + ff2_b  -> d_out
    gemm_wmma<NFF, ND, 0, 0><<<dim3(ND / 64, M / 16), 32, 0, stream>>>(ffmid, ff2_w, ff2_b, h2, out);

    // 12) gate + coord_delta + coords_out
    finalize_kernel<<<M, 256, 0, stream>>>(g1, cg2_w, cg2_b, logits, coords, coords_out);
}